// SRU_Compute_GPU_PROJ_5153960755668
// MI455X (gfx1250) — compile-verified
//
#include <hip/hip_runtime.h>

// SRU PROJ forward scan for MI455X (gfx1250, wave32).
// Memory-bound streaming recurrence: ~402 MB traffic, ~17us floor @ 23.3 TB/s.
// Strategy: 1 column/thread (1024 wave32s), 16-deep ping-pong register pipeline
// of non-temporal global_load_b64, global_prefetch_b8 staging ~16MB/0.7us ahead
// into GL2, non-temporal b32 stores. Sigmoid via raw v_exp_f32 + v_rcp_f32
// (TRANS ops co-execute with VALU) to keep VALU/element ~6 so the kernel stays
// bandwidth-bound, not issue-bound.

typedef float vfloat2 __attribute__((ext_vector_type(2)));

#define SRU_UNROLL 16   // timesteps per pipeline chunk
#define SRU_PF_CH   4   // prefetch distance in chunks (64 timesteps ahead)

#define SRU_NLOG2E (-1.4426950408889634f)

// sigmoid(x) = 1 / (1 + exp2(-x * log2(e)))
// xl2 must be  x * -log2(e)  (caller folds the bias term into an fma)
__device__ __forceinline__ float sigmoid_from_xlog2(float xl2) {
    const float e = __builtin_amdgcn_exp2f(xl2);     // v_exp_f32
    return __builtin_amdgcn_rcpf(1.0f + e);          // v_add + v_rcp_f32
}

__global__ void __launch_bounds__(128)
sru_proj_fwd_kernel(const float* __restrict__ u,     // [L, N, 2], N = B*D
                    const float* __restrict__ bias,  // [D]
                    const float* __restrict__ init,  // [N]
                    float* __restrict__ c,           // [L, N]
                    float* __restrict__ last,        // [N]
                    int L, int N, int D)
{
    const int idx = blockIdx.x * blockDim.x + threadIdx.x;
    if (idx >= N) return;

    const vfloat2* __restrict__ u2 = reinterpret_cast<const vfloat2*>(u) + idx;
    float* __restrict__ cp = c + idx;

    // Pre-fold bias into the exp2 argument:  -(u1 + bv)*log2e = u1*NLOG2E + bvl
    const float bvl = bias[idx % D] * SRU_NLOG2E;
    float cur = init[idx];

    const int nch = L / SRU_UNROLL;

    vfloat2 va[SRU_UNROLL], vb[SRU_UNROLL];

    auto load_chunk = [&](vfloat2* v, int chunk) {
        const vfloat2* p = u2 + (long)chunk * SRU_UNROLL * N;
        #pragma unroll
        for (int j = 0; j < SRU_UNROLL; ++j)
            v[j] = __builtin_nontemporal_load(p + (long)j * N);  // b64 NT
    };

    auto prefetch_chunk = [&](int chunk) {
        if (chunk < nch) {
            const vfloat2* p = u2 + (long)chunk * SRU_UNROLL * N;
            #pragma unroll
            for (int j = 0; j < SRU_UNROLL; ++j)
                __builtin_prefetch((const void*)(p + (long)j * N), 0, 2); // global_prefetch_b8
        }
    };

    auto comp_chunk = [&](vfloat2* v, int chunk) {
        float* p = cp + (long)chunk * SRU_UNROLL * N;
        #pragma unroll
        for (int j = 0; j < SRU_UNROLL; ++j) {
            const float u0 = v[j].x;
            const float g  = sigmoid_from_xlog2(
                                 __builtin_fmaf(v[j].y, SRU_NLOG2E, bvl));
            cur = __builtin_fmaf(cur - u0, g, u0);   // (cur-u0)*g + u0
            __builtin_nontemporal_store(cur, p + (long)j * N);  // b32 NT
        }
    };

    if (nch > 0) {
        load_chunk(va, 0);                 // prologue: chunk 0 in flight
        int ch = 0;
        for (; ch + 2 <= nch; ch += 2) {
            load_chunk(vb, ch + 1);        // next chunk's 16 loads in flight
            prefetch_chunk(ch + SRU_PF_CH);
            comp_chunk(va, ch);            // consume current while vb flies
            if (ch + 2 < nch) load_chunk(va, ch + 2);
            prefetch_chunk(ch + 1 + SRU_PF_CH);
            comp_chunk(vb, ch + 1);
        }
        if (ch < nch) comp_chunk(va, ch);  // odd chunk-count leftover
    }

    // tail timesteps (L not multiple of SRU_UNROLL)
    for (int t = nch * SRU_UNROLL; t < L; ++t) {
        const vfloat2 v = u2[(long)t * N];
        const float u0 = v.x;
        const float g  = sigmoid_from_xlog2(
                             __builtin_fmaf(v.y, SRU_NLOG2E, bvl));
        cur = __builtin_fmaf(cur - u0, g, u0);
        cp[(long)t * N] = cur;
    }

    last[idx] = cur;
}

extern "C" void kernel_launch(void* const* d_in, const int* in_sizes, int n_in,
                              void* d_out, int out_size, void* d_ws, size_t ws_size,
                              hipStream_t stream) {
    (void)d_ws; (void)ws_size; (void)out_size;
    if (n_in < 3) return;
    const float* u    = (const float*)d_in[0];   // [L,B,D,2]
    const float* bias = (const float*)d_in[1];   // [D]
    const float* init = (const float*)d_in[2];   // [B,D]

    const int D = in_sizes[1];
    const int N = in_sizes[2];                   // B*D columns
    if (D <= 0 || N <= 0) return;
    const int L = in_sizes[0] / (2 * N);

    float* c    = (float*)d_out;                 // [L,B,D]
    float* last = (float*)d_out + (size_t)L * N; // [B,D]

    const int block = 128;                       // 4 wave32s per block
    const int grid  = (N + block - 1) / block;   // 256 blocks -> 1024 waves
    sru_proj_fwd_kernel<<<grid, block, 0, stream>>>(u, bias, init, c, last, L, N, D);
}